// CalculatorBaseTorch_25082609008700
// MI455X (gfx1250) — compile-verified
//
#include <hip/hip_runtime.h>

// Memory-bound pairwise scatter-add for MI455X (gfx1250, wave32).
//   potential[i] += charges[j] * (0.5/d);  potential[j] += charges[i] * (0.5/d)
// HBM traffic = 96MB pair stream (NT loads); charges/potential (3.2MB each)
// stay L2-resident (192MB L2). Bottleneck: L2 atomic f32 add throughput.

typedef int   v4i __attribute__((ext_vector_type(4)));
typedef float v2f __attribute__((ext_vector_type(2)));
typedef float v4f __attribute__((ext_vector_type(4)));

__device__ __forceinline__ void atomic_add_f32(float* p, float v) {
  // Relaxed, device-scope, result unused -> global_atomic_add_f32 (no return).
  __hip_atomic_fetch_add(p, v, __ATOMIC_RELAXED, __HIP_MEMORY_SCOPE_AGENT);
}

__global__ __launch_bounds__(256) void zero_out_kernel(v4f* __restrict__ out, int n4) {
  int i = blockIdx.x * 256 + threadIdx.x;
  if (i < n4) {
    v4f z = {0.0f, 0.0f, 0.0f, 0.0f};
    out[i] = z;
  }
}

// 2 pairs per thread iteration: b128 index load + b64 distance load, both NT.
__global__ __launch_bounds__(256) void pair_scatter_kernel(
    const v4f* __restrict__ charges,    // [n_atoms] of float4 (N_CH == 4)
    const v4i* __restrict__ pair_idx2,  // [n2]: (i0, j0, i1, j1)
    const v2f* __restrict__ pair_d2,    // [n2]: (d0, d1)
    float*     __restrict__ out,        // [n_atoms * 4]
    int n2)
{
  const int stride = (int)(gridDim.x * blockDim.x);
  for (int t = blockIdx.x * 256 + threadIdx.x; t < n2; t += stride) {
    if (t + stride < n2) {
      // gfx1250 global_prefetch_b8: pull next grid-stride tile toward L2.
      __builtin_prefetch(&pair_idx2[t + stride], 0, 0);
      __builtin_prefetch(&pair_d2[t + stride], 0, 0);
    }
    // Streaming data: read once, keep out of L2 (TH=NT).
    v4i ij = __builtin_nontemporal_load(&pair_idx2[t]);
    v2f dd = __builtin_nontemporal_load(&pair_d2[t]);

#pragma unroll
    for (int k = 0; k < 2; ++k) {
      const int ai = ij[2 * k + 0];
      const int aj = ij[2 * k + 1];
      const float w = 0.5f / dd[k];        // fold the final /2 into w

      // Hot 3.2MB table: regular (RT) cached b128 loads, L2-resident.
      const v4f qi = charges[ai];
      const v4f qj = charges[aj];

      float* oi = out + 4 * ai;
      float* oj = out + 4 * aj;
#pragma unroll
      for (int c = 0; c < 4; ++c) {
        atomic_add_f32(oi + c, qj[c] * w);
        atomic_add_f32(oj + c, qi[c] * w);
      }
    }
  }
}

// Scalar tail for an odd pair count (N_PAIRS is even, but stay generic).
__global__ void pair_scatter_tail_kernel(
    const v4f* __restrict__ charges,
    const int* __restrict__ idx,
    const float* __restrict__ dist,
    float* __restrict__ out,
    int start, int n_pairs)
{
  int p = start + (int)(blockIdx.x * blockDim.x + threadIdx.x);
  if (p >= n_pairs) return;
  const int ai = idx[2 * p + 0];
  const int aj = idx[2 * p + 1];
  const float w = 0.5f / dist[p];
  const v4f qi = charges[ai];
  const v4f qj = charges[aj];
  float* oi = out + 4 * ai;
  float* oj = out + 4 * aj;
#pragma unroll
  for (int c = 0; c < 4; ++c) {
    atomic_add_f32(oi + c, qj[c] * w);
    atomic_add_f32(oj + c, qi[c] * w);
  }
}

extern "C" void kernel_launch(void* const* d_in, const int* in_sizes, int n_in,
                              void* d_out, int out_size, void* d_ws, size_t ws_size,
                              hipStream_t stream) {
  (void)n_in; (void)d_ws; (void)ws_size;

  const v4f*   charges = (const v4f*)d_in[0];   // float32 [N_ATOMS, 4]
  const int*   idx     = (const int*)d_in[1];   // int32   [N_PAIRS, 2]
  const float* dist    = (const float*)d_in[2]; // float32 [N_PAIRS]
  float*       out     = (float*)d_out;         // float32 [N_ATOMS, 4]

  const int n_pairs = in_sizes[2];

  // 1) Zero the (poisoned) output accumulator.
  const int n4 = out_size / 4;
  zero_out_kernel<<<(n4 + 255) / 256, 256, 0, stream>>>((v4f*)d_out, n4);

  // 2) Scatter. Grid-stride: ~4 iterations/thread so the prefetch has a target.
  const int n2 = n_pairs / 2;
  int blocks = (n2 + 255) / 256;
  if (blocks > 4096) blocks = 4096;
  if (blocks < 1) blocks = 1;
  if (n2 > 0) {
    pair_scatter_kernel<<<blocks, 256, 0, stream>>>(
        charges, (const v4i*)idx, (const v2f*)dist, out, n2);
  }

  // 3) Odd tail, if any.
  const int rem = n_pairs - 2 * n2;
  if (rem > 0) {
    pair_scatter_tail_kernel<<<1, 32, 0, stream>>>(charges, idx, dist, out, 2 * n2, n_pairs);
  }
}